// yoloLoss_26336739459610
// MI455X (gfx1250) — compile-verified
//
#include <hip/hip_runtime.h>

// ---------------------------------------------------------------------------
// YOLO v1 loss on gfx1250 (MI455X).
// Memory-bound: 122 MB in / 4 B out -> HBM floor ~5.2us @ 23.3 TB/s; inputs
// fit in the 192 MB L2, so default RT caching lets timed replays run from L2.
// Wave reductions use V_WMMA_F32_16X16X4_F32 (ones-matrix trick) instead of
// ds_swizzle chains.
// ---------------------------------------------------------------------------

typedef float v2f __attribute__((ext_vector_type(2)));
typedef float v8f __attribute__((ext_vector_type(8)));

#define SGRID 7
#define CELLS_PER_IMG 49
#define NBLOCKS 2048

// Exact 32-lane sum using two chained 16x16x4 f32 WMMAs.
// A = {v, 0} per lane, B = ones:
//   D[m][n] = v[m] + v[m+16]  (independent of n)  -- folds half-waves
// D layout: lane L<16 VGPR r = D[r][L]; lane L>=16 VGPR r = D[8+r][L-16],
// so an intra-lane 8-term fold gives sum(rows 0..7) on lanes 0..15 and
// sum(rows 8..15) on lanes 16..31; a second identical WMMA folds those two,
// leaving the exact total replicated in every lane.
__device__ __forceinline__ float wave_sum_wmma(float v) {
  v2f a;  a[0] = v;    a[1] = 0.0f;
  v2f b;  b[0] = 1.0f; b[1] = 1.0f;
  v8f c = {};
  c = __builtin_amdgcn_wmma_f32_16x16x4_f32(false, a, false, b,
                                            (short)0, c, false, false);
  float s = ((c[0] + c[1]) + (c[2] + c[3])) + ((c[4] + c[5]) + (c[6] + c[7]));
  v2f a2; a2[0] = s; a2[1] = 0.0f;
  v8f d = {};
  d = __builtin_amdgcn_wmma_f32_16x16x4_f32(false, a2, false, b,
                                            (short)0, d, false, false);
  return d[0];
}

__device__ __forceinline__ float block_reduce_store(float acc, float* dst) {
  float wsum = wave_sum_wmma(acc);        // all 32 lanes active (EXEC all 1s)
  __shared__ float smem[8];
  const int lane = threadIdx.x & 31;
  const int wv   = threadIdx.x >> 5;
  if (lane == 0) smem[wv] = wsum;
  __syncthreads();
  if (threadIdx.x == 0) {
    float bsum = 0.0f;
#pragma unroll
    for (int w = 0; w < 8; ++w) bsum += smem[w];
    *dst = bsum;
  }
  return 0.0f;
}

__global__ __launch_bounds__(256) void yolo_loss_partial(
    const float* __restrict__ pred, const float* __restrict__ targ,
    float* __restrict__ partial, int ncells) {
  float acc = 0.0f;
  const int stride = gridDim.x * blockDim.x;
  for (int c = blockIdx.x * blockDim.x + threadIdx.x; c < ncells; c += stride) {
    // prediction row: 48 B, 16B-aligned -> 3x global_load_b128
    const float4* p4 = reinterpret_cast<const float4*>(pred + (size_t)c * 12);
    const float4 q0 = p4[0], q1 = p4[1], q2 = p4[2];
    const float p0 = q0.x, p1 = q0.y, p2 = q0.z, p3 = q0.w;
    const float p4v = q1.x, p5 = q1.y, p6 = q1.z, p7 = q1.w;
    const float p8 = q2.x, p9 = q2.y, p10 = q2.z, p11 = q2.w;
    // target row: 28 B -> scalar dword loads (wave still touches a
    // contiguous 896 B span, full cacheline utilization)
    const float* t = targ + (size_t)c * 7;
    const float t0 = t[0], t1 = t[1], t2 = t[2], t3 = t[3];
    const float t4 = t[4], t5 = t[5], t6 = t[6];

    const int rem = c % CELLS_PER_IMG;
    const float row = (float)(rem / SGRID);
    const float col = (float)(rem % SGRID);

    // ---- target corners (match reference op order) ----
    const float tcx = (t0 + col) / 7.0f;
    const float tcy = (t1 + row) / 7.0f;
    const float tx1 = tcx - t2 * 0.5f, ty1 = tcy - t3 * 0.5f;
    const float tx2 = tcx + t2 * 0.5f, ty2 = tcy + t3 * 0.5f;
    const float ta  = fabsf((tx2 - tx1) * (ty2 - ty1));

    // ---- IoU for the two predicted boxes ----
    float iou0, iou1;
    {
      const float cx = (p0 + col) / 7.0f, cy = (p1 + row) / 7.0f;
      const float x1 = cx - p2 * 0.5f, y1 = cy - p3 * 0.5f;
      const float x2 = cx + p2 * 0.5f, y2 = cy + p3 * 0.5f;
      const float inter = fmaxf(fminf(x2, tx2) - fmaxf(x1, tx1), 0.0f) *
                          fmaxf(fminf(y2, ty2) - fmaxf(y1, ty1), 0.0f);
      const float pa = fabsf((x2 - x1) * (y2 - y1));
      iou0 = inter / (pa + ta - inter + 1e-8f);
    }
    {
      const float cx = (p5 + col) / 7.0f, cy = (p6 + row) / 7.0f;
      const float x1 = cx - p7 * 0.5f, y1 = cy - p8 * 0.5f;
      const float x2 = cx + p7 * 0.5f, y2 = cy + p8 * 0.5f;
      const float inter = fmaxf(fminf(x2, tx2) - fmaxf(x1, tx1), 0.0f) *
                          fmaxf(fminf(y2, ty2) - fmaxf(y1, ty1), 0.0f);
      const float pa = fabsf((x2 - x1) * (y2 - y1));
      iou1 = inter / (pa + ta - inter + 1e-8f);
    }

    // argmax (first max wins -> strict >)
    const bool pick1 = iou1 > iou0;
    const float sx = pick1 ? p5 : p0;
    const float sy = pick1 ? p6 : p1;
    const float sw = pick1 ? p7 : p2;
    const float sh = pick1 ? p8 : p3;
    const float sp = pick1 ? p9 : p4v;

    const float obj = (t4 > 0.0f) ? 1.0f : 0.0f;

    const float dxc = sx - t0, dyc = sy - t1;
    const float centre = dxc * dxc + dyc * dyc;

    const float sgw = (sw > 0.0f ? 1.0f : 0.0f) - (sw < 0.0f ? 1.0f : 0.0f);
    const float sgh = (sh > 0.0f ? 1.0f : 0.0f) - (sh < 0.0f ? 1.0f : 0.0f);
    const float dw = sgw * sqrtf(fabsf(sw) + 1e-6f) - sqrtf(t2);
    const float dh = sgh * sqrtf(fabsf(sh) + 1e-6f) - sqrtf(t3);
    const float dim = dw * dw + dh * dh;

    const float dcf = sp - t4;
    const float conf = dcf * dcf;

    const float n0 = p4v - t4, n1 = p9 - t4;
    const float noobj = n0 * n0 + n1 * n1;

    const float c0 = p10 - t5, c1 = p11 - t6;
    const float cls = c0 * c0 + c1 * c1;

    acc += obj * (5.0f * (centre + dim) + conf + cls) +
           (1.0f - obj) * 0.5f * noobj;
  }

  block_reduce_store(acc, &partial[blockIdx.x]);
}

__global__ __launch_bounds__(256) void yolo_loss_finish(
    const float* __restrict__ partial, int n, float* __restrict__ out) {
  float acc = 0.0f;
  for (int i = threadIdx.x; i < n; i += 256) acc += partial[i];
  block_reduce_store(acc, out);
}

extern "C" void kernel_launch(void* const* d_in, const int* in_sizes, int n_in,
                              void* d_out, int out_size, void* d_ws, size_t ws_size,
                              hipStream_t stream) {
  const float* pred = (const float*)d_in[0];
  const float* targ = (const float*)d_in[1];
  float* out = (float*)d_out;
  float* partial = (float*)d_ws;          // NBLOCKS floats = 8 KB, fully
                                          // overwritten every call
  const int ncells = in_sizes[1] / 7;     // BS * S * S

  yolo_loss_partial<<<NBLOCKS, 256, 0, stream>>>(pred, targ, partial, ncells);
  yolo_loss_finish<<<1, 256, 0, stream>>>(partial, NBLOCKS, out);
}